// GraphConvolution_26774826123836
// MI455X (gfx1250) — compile-verified
//
#include <hip/hip_runtime.h>

// Problem constants from the reference
#define N_NODES 50000   // 3125 row tiles of 16
#define DIM     128     // D_IN == D_OUT
#define E_EDGES 800000

typedef __attribute__((ext_vector_type(2))) float v2f;
typedef __attribute__((ext_vector_type(8))) float v8f;

// ---------------------------------------------------------------------------
// Kernel 1: fused dual GEMM via V_WMMA_F32_16X16X4_F32 (wave32, fp32 in/out).
//   blockIdx.x even -> support = x @ weight      (d_ws)
//   blockIdx.x odd  -> out     = x @ root_weight (d_out, initializes accum)
//
// One wave computes a 16x128 strip (8 column tiles): the A fragment loaded
// once per K-step feeds 8 WMMAs. The 64KB weight matrix is staged in LDS
// cooperatively (coalesced float4), B fragments come from ds_load_b32.
//
// Fragment layouts (CDNA5 ISA 7.12.2, 32-bit data, wave32):
//   A 16x4 (MxK):  lane l in [0,15]: M=l; VGPR v holds K = 2*(lane>>4)+v
//   B 4x16 (KxN):  lane l in [0,15]: N=l; VGPR v holds K = 2*(lane>>4)+v
//   C/D 16x16:     VGPR v: lanes 0-15 -> M=v, lanes 16-31 -> M=v+8; N=lane&15
// ---------------------------------------------------------------------------
__global__ __launch_bounds__(256) void gcn_gemm_wmma_f32(
    const float* __restrict__ x,
    const float* __restrict__ weight,
    const float* __restrict__ root_weight,
    float* __restrict__ support,
    float* __restrict__ out)
{
    __shared__ float sB[DIM * DIM];              // 64 KB LDS: one weight matrix

    const int mat = blockIdx.x & 1;              // 0: weight, 1: root_weight
    const float* __restrict__ Bg  = mat ? root_weight : weight;
    float*       __restrict__ Dst = mat ? out         : support;

    // Cooperative stage: 16384 floats / 256 threads = 16 float4 per thread.
    {
        const float4* src = (const float4*)Bg;
        float4*       dst = (float4*)sB;
#pragma unroll
        for (int i = 0; i < (DIM * DIM / 4) / 256; ++i)
            dst[threadIdx.x + i * 256] = src[threadIdx.x + i * 256];
    }
    __syncthreads();

    const int wave = threadIdx.x >> 5;
    const int lane = threadIdx.x & 31;
    const int rt   = (blockIdx.x >> 1) * 8 + wave;   // row tile for this wave
    if (rt >= N_NODES / 16) return;

    const int g = lane >> 4;     // 16-lane half select
    const int l = lane & 15;

    const float* arow = x + (size_t)(rt * 16 + l) * DIM;   // A: M = l

    v8f acc[8] = {};             // 16x128 strip: 8 column tiles
#pragma unroll 2
    for (int k0 = 0; k0 < DIM; k0 += 4) {
        const int kb = k0 + 2 * g;           // this half's K base
        v2f a;
        a.x = arow[kb + 0];                  // contiguous pair -> b64 load
        a.y = arow[kb + 1];
        const float* bp = sB + kb * DIM + l; // B[kb][l] in LDS
#pragma unroll
        for (int ct = 0; ct < 8; ++ct) {
            v2f b;
            b.x = bp[ct * 16];               // B[kb  ][ct*16+l]
            b.y = bp[ct * 16 + DIM];         // B[kb+1][ct*16+l]
            acc[ct] = __builtin_amdgcn_wmma_f32_16x16x4_f32(
                false, a, false, b, (short)0, acc[ct], false, false);
        }
    }

    // Store the 16x128 strip per the C/D layout.
    float* dbase = Dst + (size_t)(rt * 16 + 8 * g) * DIM + l;
#pragma unroll
    for (int ct = 0; ct < 8; ++ct)
#pragma unroll
        for (int v = 0; v < 8; ++v)
            dbase[(size_t)v * DIM + ct * 16] = acc[ct][v];
}

// ---------------------------------------------------------------------------
// Kernel 2: COO SpMM scatter.  out[row] += val * support[col]
// One wave per edge (32 lanes x 4 features = 128): float4 gather (support is
// 25.6MB, L2-resident vs 192MB L2) + 4x global_atomic_add_f32 into out.
// Index loads are wave-uniform -> one cache line per wave.
// ---------------------------------------------------------------------------
__global__ __launch_bounds__(256) void gcn_edge_scatter(
    const float* __restrict__ support,
    const int*   __restrict__ rows,
    const int*   __restrict__ cols,
    const float* __restrict__ vals,
    float*       __restrict__ out)
{
    const long long t     = (long long)blockIdx.x * blockDim.x + threadIdx.x;
    const long long total = (long long)E_EDGES * (DIM / 4);   // 25.6M
    if (t >= total) return;

    const int e  = (int)(t >> 5);          // 32 lanes cover one edge
    const int d4 = ((int)t & 31) * 4;

    const int   r = rows[e];
    const int   s = cols[e];
    const float v = vals[e];

    const float4 m = *(const float4*)(support + (size_t)s * DIM + d4);
    float* o = out + (size_t)r * DIM + d4;

    atomicAdd(o + 0, v * m.x);
    atomicAdd(o + 1, v * m.y);
    atomicAdd(o + 2, v * m.z);
    atomicAdd(o + 3, v * m.w);
}

// ---------------------------------------------------------------------------
// Host launcher. Input order: x, edge_rows, edge_cols, edge_vals, weight,
// root_weight. d_out: [N, DIM] fp32. d_ws holds support [N, DIM] fp32 (25.6MB).
// ---------------------------------------------------------------------------
extern "C" void kernel_launch(void* const* d_in, const int* in_sizes, int n_in,
                              void* d_out, int out_size, void* d_ws, size_t ws_size,
                              hipStream_t stream) {
    const float* x           = (const float*)d_in[0];
    const int*   edge_rows   = (const int*)  d_in[1];
    const int*   edge_cols   = (const int*)  d_in[2];
    const float* edge_vals   = (const float*)d_in[3];
    const float* weight      = (const float*)d_in[4];
    const float* root_weight = (const float*)d_in[5];

    float* out     = (float*)d_out;
    float* support = (float*)d_ws;   // N_NODES * DIM * 4 = 25.6 MB scratch

    // GEMM phase: 3125 row tiles, 8 waves (one 16x128 strip each) per block,
    // x2 blocks (even: weight, odd: root_weight).
    const int row_tile_blocks = (N_NODES / 16 + 7) / 8;    // 391
    gcn_gemm_wmma_f32<<<row_tile_blocks * 2, 256, 0, stream>>>(
        x, weight, root_weight, support, out);

    // Scatter phase (stream-ordered after GEMM initializes `out`).
    const long long threads = (long long)E_EDGES * (DIM / 4);
    const int scat_blocks   = (int)((threads + 255) / 256);
    gcn_edge_scatter<<<scat_blocks, 256, 0, stream>>>(support, edge_rows,
                                                      edge_cols, edge_vals, out);
}